// ReaxML_74431783239800
// MI455X (gfx1250) — compile-verified
//
#include <hip/hip_runtime.h>
#include <hip/hip_bf16.h>
#include <stdint.h>

#define NB 32
#define NA 1024
#define ROWS 16                  // rows per block tile
#define CCH 256                  // columns per chunk
#define RB_PER_BATCH (NA / ROWS) // 64 row-blocks per batch
#define THREADS 256
#define F4_PER_THREAD 4          // (ROWS*CCH/4) float4 / 256 threads

typedef __attribute__((ext_vector_type(2))) float v2f;
typedef __attribute__((ext_vector_type(8))) float v8f;

// CDNA5 async global->LDS copy (tracked by ASYNCcnt). vdst = LDS byte address,
// vaddr = 64-bit global address (GV mode), saddr = off.
__device__ __forceinline__ void async_load_b128(uint32_t lds_off, uint64_t gaddr) {
    asm volatile("global_load_async_to_lds_b128 %0, %1, off"
                 :: "v"(lds_off), "v"(gaddr) : "memory");
}
__device__ __forceinline__ void wait_async_le4() {
    asm volatile("s_wait_asynccnt 0x4" ::: "memory");
}
__device__ __forceinline__ void wait_async_le0() {
    asm volatile("s_wait_asynccnt 0x0" ::: "memory");
}

__global__ __launch_bounds__(THREADS)
void reax_pair_kernel(const int* __restrict__ Z, const float* __restrict__ B,
                      const float* __restrict__ d0t, const float* __restrict__ p1t,
                      const float* __restrict__ p2t, float* __restrict__ ws)
{
    __shared__ float  buf[2][ROWS * CCH];   // 2 x 16KB double buffer
    __shared__ int    zc[NA];               // atom-type index per column (and row)
    __shared__ float4 param[9];             // {d0, p1, p2, 0} flattened 3x3
    __shared__ float  red[THREADS];

    const int tid = threadIdx.x;
    const int bid = blockIdx.x;
    const int b   = bid >> 6;       // batch
    const int rb  = bid & 63;       // row-block
    const int r0  = rb << 4;

    // Stage Z->idx map for all 1024 columns (rows are a subset of these).
    const int* Zb = Z + b * NA;
#pragma unroll
    for (int k = 0; k < 4; ++k) {
        int c = tid + (k << 8);
        int z = Zb[c];
        zc[c] = (z == 1) ? 1 : ((z == 8) ? 2 : 0);
    }
    if (tid < 9) param[tid] = make_float4(d0t[tid], p1t[tid], p2t[tid], 0.0f);

    // Only columns [0, r0+ROWS) can satisfy col < row for rows in this block.
    const int nch = (r0 + ROWS + CCH - 1) >> 8;

    const uint64_t Bbase  = (uint64_t)(uintptr_t)(B + (size_t)b * NA * NA);
    const uint32_t lbase0 = (uint32_t)(uintptr_t)&buf[0][0];
    const uint32_t lbase1 = (uint32_t)(uintptr_t)&buf[1][0];

    // Issue chunk 0 into buffer 0.
#pragma unroll
    for (int k = 0; k < F4_PER_THREAD; ++k) {
        int q  = tid + (k << 8);
        int rl = q >> 6, c4 = q & 63;
        uint64_t g = Bbase + (uint64_t)(uint32_t)((((r0 + rl) << 10) + (c4 << 2)) << 2);
        async_load_b128(lbase0 + (q << 4), g);
    }
    __syncthreads();   // zc/param visible to all waves

    float acc = 0.0f;
    for (int ch = 0; ch < nch; ++ch) {
        const int nb = ch & 1;
        if (ch + 1 < nch) {
            // Prefetch next chunk into the other buffer, then wait for current.
            const int c0n = (ch + 1) << 8;
            const uint32_t lb = nb ? lbase0 : lbase1;
#pragma unroll
            for (int k = 0; k < F4_PER_THREAD; ++k) {
                int q  = tid + (k << 8);
                int rl = q >> 6, c4 = q & 63;
                uint64_t g = Bbase + (uint64_t)(uint32_t)((((r0 + rl) << 10) + c0n + (c4 << 2)) << 2);
                async_load_b128(lb + (q << 4), g);
            }
            wait_async_le4();   // <=4 outstanding -> current chunk's 4 are done
        } else {
            wait_async_le0();
        }
        __syncthreads();        // all waves' async data for this chunk landed

        const int c0 = ch << 8;
        const float4* tile = (const float4*)buf[nb];
#pragma unroll
        for (int k = 0; k < F4_PER_THREAD; ++k) {
            int q  = tid + (k << 8);
            int rl = q >> 6, c4 = q & 63;
            int gr = r0 + rl;
            int ir3 = zc[gr] * 3;
            float4 v = tile[q];
            int cb = c0 + (c4 << 2);
            int4 zz = *(const int4*)&zc[cb];
            {
                float x = fabsf(v.x);
                float4 pm = param[ir3 + zz.x];
                float e = pm.x * x * __expf(pm.y * (1.0f - __powf(x, pm.z)));
                acc += (cb + 0 < gr) ? e : 0.0f;
            }
            {
                float x = fabsf(v.y);
                float4 pm = param[ir3 + zz.y];
                float e = pm.x * x * __expf(pm.y * (1.0f - __powf(x, pm.z)));
                acc += (cb + 1 < gr) ? e : 0.0f;
            }
            {
                float x = fabsf(v.z);
                float4 pm = param[ir3 + zz.z];
                float e = pm.x * x * __expf(pm.y * (1.0f - __powf(x, pm.z)));
                acc += (cb + 2 < gr) ? e : 0.0f;
            }
            {
                float x = fabsf(v.w);
                float4 pm = param[ir3 + zz.w];
                float e = pm.x * x * __expf(pm.y * (1.0f - __powf(x, pm.z)));
                acc += (cb + 3 < gr) ? e : 0.0f;
            }
        }
        __syncthreads();        // buffer free before it is overwritten next iter
    }

    red[tid] = acc;
    __syncthreads();

    // Block reduction: wave 0 folds 256 -> 32 partials, then a 5-WMMA ones-matrix
    // reduction replicates the grand total into every element of D.
    if (tid < 32) {             // whole wave 0 active -> EXEC all-ones for WMMA
        float partial = 0.0f;
#pragma unroll
        for (int k = 0; k < 8; ++k) partial += red[tid + (k << 5)];

        v2f a;    a[0] = partial; a[1] = 0.0f;   // A(16x4): K=0/2 = partials, K=1/3 = 0
        v2f ones; ones[0] = 1.0f; ones[1] = 1.0f;
        v8f c1 = {};
        // D1[m][n] = partial[m] + partial[m+16]
        c1 = __builtin_amdgcn_wmma_f32_16x16x4_f32(false, a, false, ones,
                                                   (short)0, c1, false, false);
        v8f c2 = {};
        // Feed D1's VGPR pairs back as B with all-ones A: after 4 accumulating
        // WMMAs every entry of D2 equals sum over all 16 rows = total.
#pragma unroll
        for (int p = 0; p < 4; ++p) {
            v2f bp; bp[0] = c1[2 * p]; bp[1] = c1[2 * p + 1];
            c2 = __builtin_amdgcn_wmma_f32_16x16x4_f32(false, ones, false, bp,
                                                       (short)0, c2, false, false);
        }
        if (tid == 0) ws[bid] = -c2[0];   // e_ij carries the minus sign
    }
}

__global__ __launch_bounds__(64)
void reax_reduce_kernel(const float* __restrict__ ws, float* __restrict__ out)
{
    const int b = blockIdx.x;
    const int t = threadIdx.x;
    float v = ws[b * RB_PER_BATCH + t];
#pragma unroll
    for (int off = 16; off >= 1; off >>= 1) v += __shfl_down(v, off, 32);
    __shared__ float s2[2];
    if ((t & 31) == 0) s2[t >> 5] = v;
    __syncthreads();
    if (t == 0) out[b] = s2[0] + s2[1];
}

extern "C" void kernel_launch(void* const* d_in, const int* in_sizes, int n_in,
                              void* d_out, int out_size, void* d_ws, size_t ws_size,
                              hipStream_t stream) {
    (void)in_sizes; (void)n_in; (void)out_size; (void)ws_size;
    const int*   Z  = (const int*)d_in[0];
    const float* B  = (const float*)d_in[1];
    const float* d0 = (const float*)d_in[2];
    const float* p1 = (const float*)d_in[3];
    const float* p2 = (const float*)d_in[4];
    float* ws  = (float*)d_ws;   // 2048 block partials (8 KB)
    float* out = (float*)d_out;  // 32 batch energies

    reax_pair_kernel<<<dim3(NB * RB_PER_BATCH), dim3(THREADS), 0, stream>>>(
        Z, B, d0, p1, p2, ws);
    reax_reduce_kernel<<<dim3(NB), dim3(64), 0, stream>>>(ws, out);
}